// SparseGlobalBroadcast_54726473285929
// MI455X (gfx1250) — compile-verified
//
#include <hip/hip_runtime.h>

// SparseGlobalBroadcast (ADDITION): out[i,:] = feats[i,:] + table[bidx[i],:]
// N=1e6, C=128, B=8, fp32. Pure streaming op: ~1.03 GB moved -> ~44us floor
// at 23.3 TB/s HBM. Strategy: wave-per-row b128 streaming with NT hints,
// 4KB gather table staged into LDS (TDM async-tensor path + deterministic
// overwrite), grid-stride persistent waves for memory-level parallelism.

typedef __attribute__((ext_vector_type(4))) float        floatx4;
typedef __attribute__((ext_vector_type(4))) unsigned int uintx4;
typedef __attribute__((ext_vector_type(8))) int          intx8;
typedef __attribute__((ext_vector_type(4))) int          intx4;

#define C_DIM 128
#define B_DIM 8
#define TAB_ELEMS (B_DIM * C_DIM)   // 1024 floats = 4 KB

__global__ __launch_bounds__(256) void sgb_add_kernel(
    const float* __restrict__ feats,
    const float* __restrict__ glob,
    const int*   __restrict__ bidx,
    float*       __restrict__ out,
    int n_points)
{
  __shared__ float tab[TAB_ELEMS];

#if defined(__gfx1250__)
  // ---- CDNA5 Tensor Data Mover: async-stage the 8x128 f32 table to LDS ----
  // D# per CDNA5 ISA: 1 row x 1024 elems x 4B = 4096 B.
  if (threadIdx.x == 0) {
    unsigned long long ga = (unsigned long long)(uintptr_t)glob;
    // low 32 bits of a flat LDS pointer are the LDS byte offset
    unsigned lds_off = (unsigned)(unsigned long long)(uintptr_t)&tab[0];

    uintx4 g0;
    g0[0] = 1u;                                              // count=1, user mode
    g0[1] = lds_off;                                         // lds_addr
    g0[2] = (unsigned)(ga & 0xFFFFFFFFull);                  // global_addr[31:0]
    g0[3] = (unsigned)((ga >> 32) & 0x01FFFFFFull)           // global_addr[56:32]
            | (2u << 30);                                    // type=2 ("image")

    intx8 g1;
    g1[0] = (int)(2u << 16);                                 // data_size=4B, wg_mask=0
    g1[1] = (int)(((unsigned)TAB_ELEMS & 0xFFFFu) << 16);    // tensor_dim0 lo16 = 1024
    g1[2] = (int)(1u << 16);                                 // tensor_dim0 hi=0, tensor_dim1=1
    g1[3] = (int)(((unsigned)TAB_ELEMS & 0xFFFFu) << 16);    // tile_dim0 = 1024
    g1[4] = 1;                                               // tile_dim1=1, tile_dim2=0
    g1[5] = TAB_ELEMS;                                       // tensor_dim0_stride = 1024
    g1[6] = 0;
    g1[7] = 0;

    intx4 z4 = (intx4){0, 0, 0, 0};                          // groups 2/3 unused (2D)
    intx8 z8 = (intx8){0, 0, 0, 0, 0, 0, 0, 0};              // extra group (6-arg form)
    __builtin_amdgcn_tensor_load_to_lds(g0, g1, z4, z4, z8, 0);
  }
  __builtin_amdgcn_s_wait_tensorcnt(0);                      // no-op for non-issuing waves
  __syncthreads();
#endif

  // Deterministic staging: overwrites TDM result with identical data, so
  // correctness does not hinge on D# field semantics. 256 threads x float4
  // covers the 1024-float table exactly.
  ((floatx4*)tab)[threadIdx.x] = ((const floatx4*)glob)[threadIdx.x];
  __syncthreads();

  // ---- main stream: one wave per point row (32 lanes x 16B = 512B row) ----
  const unsigned lane = threadIdx.x & 31u;
  const unsigned wpb  = blockDim.x >> 5;
  size_t       row    = (size_t)blockIdx.x * wpb + (threadIdx.x >> 5);
  const size_t stride = (size_t)gridDim.x * wpb;
  const size_t n      = (size_t)n_points;
  const unsigned c4   = C_DIM / 4;   // 32 float4 chunks per row

  const floatx4* __restrict__ f4 = (const floatx4*)feats;
  floatx4*       __restrict__ o4 = (floatx4*)out;
  const floatx4* t4 = (const floatx4*)tab;

  for (; row < n; row += stride) {
    size_t nx = row + stride;
    if (nx < n) {
      // counter-free global_prefetch_b8 of the next row this wave will touch
      __builtin_prefetch(&f4[nx * c4 + lane], 0, 1);
    }

    int b = bidx[row];                 // wave-uniform (one row per wave)
#if defined(__AMDGCN__)
    b = __builtin_amdgcn_readfirstlane(b);
#endif

    floatx4 v = __builtin_nontemporal_load(&f4[row * c4 + lane]);  // stream-in, TH=NT
    floatx4 g = t4[(unsigned)b * c4 + lane];                       // ds_load_b128
    __builtin_nontemporal_store(v + g, &o4[row * c4 + lane]);      // stream-out, TH=NT
  }
}

extern "C" void kernel_launch(void* const* d_in, const int* in_sizes, int n_in,
                              void* d_out, int out_size, void* d_ws, size_t ws_size,
                              hipStream_t stream) {
  const float* feats = (const float*)d_in[0];   // [N, 128] f32
  const float* glob  = (const float*)d_in[1];   // [8, 128] f32
  const int*   bidx  = (const int*)d_in[2];     // [N] i32
  float*       out   = (float*)d_out;           // [N, 128] f32

  const int n = in_sizes[2];                    // number of points

  // 8 waves/block; cap blocks so every wave gets >= 1 row. 4096 blocks x 8
  // waves x 512B = 16 MB in flight, enough to cover HBM latency at 23.3 TB/s.
  int blocks = 4096;
  int max_blocks = (n + 7) / 8;
  if (blocks > max_blocks) blocks = max_blocks;
  if (blocks < 1) blocks = 1;

  sgb_add_kernel<<<blocks, 256, 0, stream>>>(feats, glob, bidx, out, n);
}